// GraphConvolution_54468775248495
// MI455X (gfx1250) — compile-verified
//
#include <hip/hip_runtime.h>

#define N_NODES 50000
#define N_EDGES 600000
#define D 128            // D_IN == D_OUT == 128
#define LDS_PITCH 132    // 128 + 4 pad: A-fragment rows land on distinct LDS banks

typedef __attribute__((ext_vector_type(2))) float v2f;
typedef __attribute__((ext_vector_type(8))) float v8f;

// ---------------------------------------------------------------------------
// Phase 1a: in-degree counting. deg[dst[e]] += 1 (native u32 atomic).
// ---------------------------------------------------------------------------
__global__ __launch_bounds__(256) void gcn_deg_kernel(
    const int* __restrict__ dst, unsigned* __restrict__ deg) {
  int e = blockIdx.x * 256 + threadIdx.x;
  if (e < N_EDGES) {
    atomicAdd(&deg[dst[e]], 1u);
  }
}

// ---------------------------------------------------------------------------
// Phase 1b: per-edge norm = rsqrt(deg[src] * deg[dst]). deg is L2-resident.
// ---------------------------------------------------------------------------
__global__ __launch_bounds__(256) void gcn_norm_kernel(
    const int* __restrict__ src, const int* __restrict__ dst,
    const unsigned* __restrict__ deg, float* __restrict__ norm) {
  int e = blockIdx.x * 256 + threadIdx.x;
  if (e < N_EDGES) {
    float p = (float)deg[src[e]] * (float)deg[dst[e]];
    norm[e] = rsqrtf(p);
  }
}

// ---------------------------------------------------------------------------
// Phase 1c: agg[dst] += x[src] * norm.  One wave32 per edge: lane handles 4
// contiguous channels (float4 gather from x, 4 coalesced f32 atomics to agg).
// x and agg both fit in the 192MB L2, so this is on-chip traffic.
// ---------------------------------------------------------------------------
__global__ __launch_bounds__(256) void gcn_scatter_kernel(
    const float* __restrict__ x, const int* __restrict__ src,
    const int* __restrict__ dst, const float* __restrict__ norm,
    float* __restrict__ agg) {
  int tid  = blockIdx.x * 256 + threadIdx.x;
  int e    = tid >> 5;        // one wave per edge
  int lane = tid & 31;        // 32 lanes x 4 channels = 128
  if (e < N_EDGES) {
    int s = src[e];
    int d = dst[e];
    float nv = norm[e];
    const float4 xv = ((const float4*)(x + (size_t)s * D))[lane];
    float* ap = agg + (size_t)d * D + lane * 4;
    __hip_atomic_fetch_add(ap + 0, xv.x * nv, __ATOMIC_RELAXED, __HIP_MEMORY_SCOPE_AGENT);
    __hip_atomic_fetch_add(ap + 1, xv.y * nv, __ATOMIC_RELAXED, __HIP_MEMORY_SCOPE_AGENT);
    __hip_atomic_fetch_add(ap + 2, xv.z * nv, __ATOMIC_RELAXED, __HIP_MEMORY_SCOPE_AGENT);
    __hip_atomic_fetch_add(ap + 3, xv.w * nv, __ATOMIC_RELAXED, __HIP_MEMORY_SCOPE_AGENT);
  }
}

// ---------------------------------------------------------------------------
// Phase 2: out = agg @ W + deg[:,None]*b, via V_WMMA_F32_16X16X4_F32.
// Block = 256 threads = 8 waves; block owns a 16-row tile of agg (staged in
// LDS, pitch-padded), wave w owns output columns [16w, 16w+16).
// Fragment layouts per CDNA5 ISA 32-bit tables:
//   A (16x4):  lane l, vgpr v -> M = l%16, K = 2*(l/16) + v
//   B (4x16):  lane l, vgpr v -> K = 2*(l/16) + v, N = l%16
//   C/D (16x16): lane l, vgpr v -> M = v + 8*(l/16), N = l%16
// ---------------------------------------------------------------------------
__global__ __launch_bounds__(256) void gcn_gemm_kernel(
    const float* __restrict__ agg, const float* __restrict__ W,
    const float* __restrict__ b, const unsigned* __restrict__ deg,
    float* __restrict__ out) {
  __shared__ float lds[16 * LDS_PITCH];

  const int t     = threadIdx.x;
  const int mbase = blockIdx.x * 16;

  // Cooperative load of the 16x128 A tile into LDS (coalesced global reads).
#pragma unroll
  for (int i = 0; i < 8; ++i) {
    int idx = t + i * 256;
    int row = idx >> 7;     // 0..15
    int col = idx & 127;    // 0..127
    lds[row * LDS_PITCH + col] = agg[(size_t)(mbase + row) * D + col];
  }
  __syncthreads();

  const int wave = t >> 5;        // 0..7  -> output column tile
  const int lane = t & 31;
  const int hi   = lane >> 4;     // 0 or 1
  const int lm   = lane & 15;
  const int col  = wave * 16 + lm;

  v8f c = {0.f, 0.f, 0.f, 0.f, 0.f, 0.f, 0.f, 0.f};
  const float* lrow = lds + lm * LDS_PITCH;   // A row for this lane (M = lm)

#pragma unroll
  for (int kk = 0; kk < 32; ++kk) {
    const int kb = kk * 4 + 2 * hi;           // K of vgpr0 element
    // A fragment: ds_load_b64 (8B aligned: pitch even, kb even)
    v2f a = *(const v2f*)(lrow + kb);
    // B fragment: two rows of W, same column
    v2f bb;
    bb.x = W[(size_t)kb * D + col];
    bb.y = W[(size_t)(kb + 1) * D + col];
    c = __builtin_amdgcn_wmma_f32_16x16x4_f32(
        /*neg_a=*/false, a, /*neg_b=*/false, bb,
        /*c_mod=*/(short)0, c, /*reuse_a=*/false, /*reuse_b=*/false);
  }

  // Bias: reference adds b once per incoming edge => + deg[row]*b[col].
  const float bcol = b[col];
#pragma unroll
  for (int v = 0; v < 8; ++v) {
    int row = mbase + v + 8 * hi;
    float o = c[v] + (float)deg[row] * bcol;
    out[(size_t)row * D + col] = o;
  }
}

// ---------------------------------------------------------------------------
// Workspace layout:
//   [0, 25.6MB)          agg  : float[N_NODES*D]
//   [25.6MB, 25.8MB)     deg  : uint [N_NODES]
//   [25.8MB, 28.2MB)     norm : float[N_EDGES]
// ---------------------------------------------------------------------------
#define AGG_BYTES  (size_t)(N_NODES * D * 4)      // 25,600,000
#define DEG_BYTES  (size_t)(N_NODES * 4)          //    200,000
#define NORM_OFF   (AGG_BYTES + DEG_BYTES)

extern "C" void kernel_launch(void* const* d_in, const int* in_sizes, int n_in,
                              void* d_out, int out_size, void* d_ws, size_t ws_size,
                              hipStream_t stream) {
  const float* x   = (const float*)d_in[0];
  const int*   src = (const int*)d_in[1];
  const int*   dst = (const int*)d_in[2];
  const float* W   = (const float*)d_in[3];
  const float* b   = (const float*)d_in[4];
  float*       out = (float*)d_out;

  char* ws = (char*)d_ws;
  float*    agg  = (float*)ws;
  unsigned* deg  = (unsigned*)(ws + AGG_BYTES);
  float*    norm = (float*)(ws + NORM_OFF);

  // Zero agg + deg each call (harness does not re-poison between replays).
  hipMemsetAsync(d_ws, 0, AGG_BYTES + DEG_BYTES, stream);

  const int eb = (N_EDGES + 255) / 256;                 // 2344 blocks
  gcn_deg_kernel<<<eb, 256, 0, stream>>>(dst, deg);
  gcn_norm_kernel<<<eb, 256, 0, stream>>>(src, dst, deg, norm);
  gcn_scatter_kernel<<<(N_EDGES * 32) / 256, 256, 0, stream>>>(x, src, dst, norm, agg);
  gcn_gemm_kernel<<<N_NODES / 16, 256, 0, stream>>>(agg, W, b, deg, out);
}